// ODEHypernet_76596446757417
// MI455X (gfx1250) — compile-verified
//
#include <hip/hip_runtime.h>
#include <hip/hip_bf16.h>

typedef __attribute__((ext_vector_type(2))) float v2f;
typedef __attribute__((ext_vector_type(8))) float v8f;

#define BATCH        64
#define NPTS         8192
#define TOTW         9356      // 448 + 4352 + 4352 + 204
#define PTS_PER_BLK  64
#define ACT_STRIDE   65        // 64 + 1 pad -> conflict-free column reads

#define LOG2E_F 1.4426950408889634f
#define LN2_F   0.6931471805599453f

// weight-blob offsets (floats) inside one batch row of target_networks_weights
#define W0_OFF   0      // 3x64
#define V0_OFF   192    // bias / w_scale / b_scale / w_shift, each 64
#define W1_OFF   448    // 64x64
#define V1_OFF   4544
#define W2_OFF   4800   // 64x64
#define V2_OFF   8896
#define W3_OFF   9152   // 64x3
#define B3_OFF   9344   // bias(3), w_scale(3), b_scale(3), w_shift(3)

// Raw-hardware transcendentals: v_exp_f32 / v_log_f32 / v_rcp_f32, no libm guards.
__device__ __forceinline__ float sigmoid_f(float x) {
    return __builtin_amdgcn_rcpf(1.0f + __builtin_amdgcn_exp2f(x * -LOG2E_F));
}
__device__ __forceinline__ float softplus_f(float x) {
    // log(1+exp(x)) = max(x,0) + ln2 * log2(1 + 2^(-|x|*log2e));  arg of log in (1,2]
    const float e = __builtin_amdgcn_exp2f(fabsf(x) * -LOG2E_F);
    return fmaf(LN2_F, __builtin_amdgcn_logf(1.0f + e), fmaxf(x, 0.0f));
}

// One 64->64 layer for one wave's 16-point tile, fp32 WMMA, in-place in LDS.
// K-outer / tile-inner: 4 independent WMMA accumulation chains for ILP.
__device__ __forceinline__ void mid_layer(float* __restrict__ act,
                                          const float* __restrict__ w_lds,
                                          int wOff, int vOff,
                                          const float ctx8[8],
                                          int lrow, bool hi)
{
    // Capture ALL A fragments first so the in-place store below is safe.
    // A 16x4 f32 layout: v0 = (M=lane, K=k0 | K=k0+2), v1 = (K=k0+1 | K=k0+3)
    v2f af[16];
    const int koff = hi ? 2 : 0;
#pragma unroll
    for (int k = 0; k < 16; ++k) {
        const float* ap = act + lrow * ACT_STRIDE + 4 * k + koff;
        af[k].x = ap[0];
        af[k].y = ap[1];
    }

    v8f acc[4] = {{}, {}, {}, {}};
#pragma unroll
    for (int k = 0; k < 16; ++k) {
        const int kb = 4 * k + koff;
        const float* wr0 = w_lds + wOff + (kb + 0) * 64 + lrow;
        const float* wr1 = w_lds + wOff + (kb + 1) * 64 + lrow;
#pragma unroll
        for (int t = 0; t < 4; ++t) {                 // 4 independent chains
            v2f bf;                                   // B 4x16: rows striped over lanes
            bf.x = wr0[t * 16];
            bf.y = wr1[t * 16];
            acc[t] = __builtin_amdgcn_wmma_f32_16x16x4_f32(
                false, af[k], false, bf, (short)0, acc[t], false, false);
        }
    }

#pragma unroll
    for (int t = 0; t < 4; ++t) {
        const int n = t * 16 + lrow;                  // output feature this lane owns
        const float bb = w_lds[vOff + n];
        const float ws = w_lds[vOff + 64 + n];
        const float bs = w_lds[vOff + 128 + n];
        const float wh = w_lds[vOff + 192 + n];
#pragma unroll
        for (int r = 0; r < 8; ++r) {                 // D: M = r + (hi?8:0), N = n
            const int m = r + (hi ? 8 : 0);
            const float c = ctx8[r];
            float v = acc[t][r] + bb;
            v = sigmoid_f(fmaf(c, ws, bs)) * v + c * wh;
            act[m * ACT_STRIDE + n] = softplus_f(v);
        }
    }
}

__global__ __launch_bounds__(128) void ode_hypernet_kernel(
    const float* __restrict__ context,   // (B, N)
    const float* __restrict__ y,         // (B, N, 3)
    const float* __restrict__ tnw,       // (B, TOTW)
    float* __restrict__ out)             // (B, N, 3)
{
    __shared__ __align__(16) float w_lds[TOTW + 4];
    __shared__ float ctx_lds[PTS_PER_BLK];
    __shared__ float act_lds[4 * 16 * ACT_STRIDE];

    const int tid  = threadIdx.x;
    const int b    = blockIdx.y;
    const int wid  = tid >> 5;
    const int lane = tid & 31;
    const int lrow = lane & 15;
    const bool hi  = lane >= 16;
    const int point0_blk = blockIdx.x * PTS_PER_BLK;

    // ---- stage this batch's weight blob into LDS (float4, fully coalesced)
    {
        const float4* src = (const float4*)(tnw + (size_t)b * TOTW);
        __builtin_prefetch(src, 0, 0);               // global_prefetch_b8
        float4* dst = (float4*)w_lds;
        for (int i = tid; i < TOTW / 4; i += 128) dst[i] = src[i];
    }
    if (tid < PTS_PER_BLK)
        ctx_lds[tid] = context[(size_t)b * NPTS + point0_blk + tid];
    __syncthreads();

    const int point0 = point0_blk + wid * 16;        // this wave's 16 points
    float* act = act_lds + wid * 16 * ACT_STRIDE;    // wave-private activations

    // per-lane context values for the 8 D-matrix rows this lane covers
    float ctx8[8];
#pragma unroll
    for (int r = 0; r < 8; ++r)
        ctx8[r] = ctx_lds[wid * 16 + r + (hi ? 8 : 0)];

    // ================= layer 0: 3 -> 64 (K padded to 4, single WMMA step)
    {
        v2f a0;
        {   // branchless: always load a valid address, mask the K=3 pad lane
            const float* yp = y + ((size_t)b * NPTS + point0 + lrow) * 3;
            const float ya = yp[hi ? 2 : 0];         // K=0 (lo) / K=2 (hi)
            const float yb = yp[1];                  // K=1
            a0.x = ya;
            a0.y = hi ? 0.0f : yb;                   // K=3 pad for hi half
        }
#pragma unroll
        for (int t = 0; t < 4; ++t) {
            const int n = t * 16 + lrow;
            v2f bf;                                   // rows K=0,1 (lo) / K=2,pad (hi)
            bf.x = w_lds[W0_OFF + (hi ? 2 : 0) * 64 + n];
            const float w1 = w_lds[W0_OFF + 1 * 64 + n];
            bf.y = hi ? 0.0f : w1;
            v8f acc = {};
            acc = __builtin_amdgcn_wmma_f32_16x16x4_f32(
                false, a0, false, bf, (short)0, acc, false, false);
            const float bb = w_lds[V0_OFF + n];
            const float ws = w_lds[V0_OFF + 64 + n];
            const float bs = w_lds[V0_OFF + 128 + n];
            const float wh = w_lds[V0_OFF + 192 + n];
#pragma unroll
            for (int r = 0; r < 8; ++r) {
                const int m = r + (hi ? 8 : 0);
                const float c = ctx8[r];
                float v = acc[r] + bb;
                v = sigmoid_f(fmaf(c, ws, bs)) * v + c * wh;
                act[m * ACT_STRIDE + n] = softplus_f(v);
            }
        }
    }

    // ================= layers 1,2: 64 -> 64
    mid_layer(act, w_lds, W1_OFF, V1_OFF, ctx8, lrow, hi);
    mid_layer(act, w_lds, W2_OFF, V2_OFF, ctx8, lrow, hi);

    // ================= layer 3: 64 -> 3 (single N tile, columns >=3 masked)
    {
        v2f af[16];
        const int koff = hi ? 2 : 0;
#pragma unroll
        for (int k = 0; k < 16; ++k) {
            const float* ap = act + lrow * ACT_STRIDE + 4 * k + koff;
            af[k].x = ap[0];
            af[k].y = ap[1];
        }
        const bool nvalid = lrow < 3;
        const int nc = nvalid ? lrow : 0;
        const float nmask = nvalid ? 1.0f : 0.0f;
        v8f acc0 = {}, acc1 = {};                     // two chains for ILP
#pragma unroll
        for (int k = 0; k < 16; k += 2) {
            const int kb0 = 4 * k + koff;
            const int kb1 = 4 * (k + 1) + koff;
            v2f bf0, bf1;
            bf0.x = w_lds[W3_OFF + (kb0 + 0) * 3 + nc] * nmask;
            bf0.y = w_lds[W3_OFF + (kb0 + 1) * 3 + nc] * nmask;
            bf1.x = w_lds[W3_OFF + (kb1 + 0) * 3 + nc] * nmask;
            bf1.y = w_lds[W3_OFF + (kb1 + 1) * 3 + nc] * nmask;
            acc0 = __builtin_amdgcn_wmma_f32_16x16x4_f32(
                false, af[k], false, bf0, (short)0, acc0, false, false);
            acc1 = __builtin_amdgcn_wmma_f32_16x16x4_f32(
                false, af[k + 1], false, bf1, (short)0, acc1, false, false);
        }
        const v8f acc = acc0 + acc1;
        if (nvalid) {
            const float bb = w_lds[B3_OFF + lrow];
            const float ws = w_lds[B3_OFF + 3 + lrow];
            const float bs = w_lds[B3_OFF + 6 + lrow];
            const float wh = w_lds[B3_OFF + 9 + lrow];
#pragma unroll
            for (int r = 0; r < 8; ++r) {
                const int m = r + (hi ? 8 : 0);
                const float c = ctx8[r];
                float v = acc[r] + bb;
                v = sigmoid_f(fmaf(c, ws, bs)) * v + c * wh;  // no softplus on last layer
                out[((size_t)b * NPTS + point0 + m) * 3 + lrow] = v;
            }
        }
    }
}

extern "C" void kernel_launch(void* const* d_in, const int* in_sizes, int n_in,
                              void* d_out, int out_size, void* d_ws, size_t ws_size,
                              hipStream_t stream) {
    const float* context = (const float*)d_in[0];
    const float* y       = (const float*)d_in[1];
    const float* tnw     = (const float*)d_in[2];
    float* out           = (float*)d_out;

    dim3 grid(NPTS / PTS_PER_BLK, BATCH);   // (128, 64) blocks
    ode_hypernet_kernel<<<grid, 128, 0, stream>>>(context, y, tnw, out);
}